// GNNML1_38422777430259
// MI455X (gfx1250) — compile-verified
//
#include <hip/hip_runtime.h>

#define NN 50000
#define DD 128
#define EE 640000

typedef __attribute__((ext_vector_type(2))) float v2f;
typedef __attribute__((ext_vector_type(8))) float v8f;

// ---------------------------------------------------------------------------
// Zero-fill (float4 granularity) for the aggregation buffer.
// ---------------------------------------------------------------------------
__global__ void __launch_bounds__(256) zero_f4(float4* __restrict__ p, int n4) {
    int i = blockIdx.x * blockDim.x + threadIdx.x;
    if (i < n4) p[i] = make_float4(0.f, 0.f, 0.f, 0.f);
}

// ---------------------------------------------------------------------------
// Pack a 128x128 weight into WMMA-B fragment order:
//   Wp(float2)[k2*128 + n] = { W[2*k2][n], W[2*k2+1][n] },  k2 = 0..63
// so each lane's B fragment {B[ka][n], B[ka+1][n]} is one aligned b64 load.
// ---------------------------------------------------------------------------
__global__ void __launch_bounds__(256) pack_weight(const float* __restrict__ w,
                                                   float* __restrict__ wp) {
    int i = blockIdx.x * blockDim.x + threadIdx.x;   // 0 .. 64*128-1
    if (i >= 64 * DD) return;
    int k2 = i >> 7, n = i & 127;
    wp[2 * i + 0] = w[(2 * k2) * DD + n];
    wp[2 * i + 1] = w[(2 * k2 + 1) * DD + n];
}

// ---------------------------------------------------------------------------
// Edge scatter: one wave (32 lanes) per edge, float4 per lane (D=128).
// h_agg[dst] += edge_attr[e] * x[src]   via global_atomic_add_f32.
// ---------------------------------------------------------------------------
__global__ void __launch_bounds__(256) edge_scatter(
    const float* __restrict__ x, const int* __restrict__ ei,
    const float* __restrict__ ea, float* __restrict__ hagg) {
    int t = blockIdx.x * blockDim.x + threadIdx.x;
    int e = t >> 5;
    if (e >= EE) return;
    int lane = t & 31;
    int src = ei[e];        // edge_index[0][e]
    int dst = ei[EE + e];   // edge_index[1][e]
    float w = ea[e];
    float4 v = ((const float4*)(x + (size_t)src * DD))[lane];
    float* o = hagg + (size_t)dst * DD + lane * 4;
    atomicAdd(o + 0, v.x * w);
    atomicAdd(o + 1, v.y * w);
    atomicAdd(o + 2, v.z * w);
    atomicAdd(o + 3, v.w * w);
}

// ---------------------------------------------------------------------------
// Fused layer: xout = relu( xin@Wl + bl + hagg@Wg + bg
//                           + (xin@Wm + bm) * (xin@Wn + bn) )
// Block = 32 rows x 128 cols (2 M-tiles of 16); 8 waves, wave w owns columns
// [16w,16w+16). B fragments (packed) are reused across both M-tiles.
// K accumulated in steps of 4 with V_WMMA_F32_16X16X4_F32 (full fp32).
//
// Fragment layouts (CDNA5 ISA 7.12.2, wave32):
//   A 16x4 f32 : lane&15 = M, kh=lane>>4; {a.x,a.y} = A[m][k+2kh .. k+2kh+1]
//   B 4x16 f32 : lane&15 = N;  {b.x,b.y} = B[k+2kh][n], B[k+2kh+1][n]
//   C 16x16 f32: VGPR i -> row i (lanes 0-15) / row i+8 (lanes 16-31)
// ---------------------------------------------------------------------------
__global__ void __launch_bounds__(256) fused_layer(
    const float* __restrict__ xin, const float* __restrict__ hagg,
    const float* __restrict__ wlp, const float* __restrict__ bl,
    const float* __restrict__ wgp, const float* __restrict__ bg,
    const float* __restrict__ wmp, const float* __restrict__ bm,
    const float* __restrict__ wnp, const float* __restrict__ bn,
    float* __restrict__ xout) {
    const int lane = threadIdx.x & 31;
    const int wave = threadIdx.x >> 5;
    const int m    = lane & 15;
    const int kh   = lane >> 4;        // 0/1: K-half of the 4-slab
    const int n    = (wave << 4) | m;  // output column 0..127
    const int row0 = blockIdx.x << 5;  // 32 rows per block

    // Tile-1 row loads clamped so EXEC stays uniform (WMMA requires all-1s);
    // garbage accumulators of an invalid tile are simply never stored.
    const int r0 = row0 + m;
    const int r1 = (row0 + 16 + m < NN) ? (row0 + 16 + m) : (NN - 1);
    const float* xr0 = xin  + (size_t)r0 * DD;
    const float* hr0 = hagg + (size_t)r0 * DD;
    const float* xr1 = xin  + (size_t)r1 * DD;
    const float* hr1 = hagg + (size_t)r1 * DD;

    const v2f* WL = (const v2f*)wlp;
    const v2f* WG = (const v2f*)wgp;
    const v2f* WM = (const v2f*)wmp;
    const v2f* WN = (const v2f*)wnp;

    v8f cl0 = {0.f,0.f,0.f,0.f,0.f,0.f,0.f,0.f};
    v8f cg0 = cl0, cm0 = cl0, cn0 = cl0;
    v8f cl1 = cl0, cg1 = cl0, cm1 = cl0, cn1 = cl0;

#pragma unroll 2
    for (int k = 0; k < DD; k += 4) {
        const int ka = k + 2 * kh;               // per-lane A/B row index
        const int kb = (k >> 1) + kh;            // packed-B fragment index
        v2f ax0 = *(const v2f*)(xr0 + ka);
        v2f ah0 = *(const v2f*)(hr0 + ka);
        v2f ax1 = *(const v2f*)(xr1 + ka);
        v2f ah1 = *(const v2f*)(hr1 + ka);
        v2f bL = WL[kb * DD + n];
        v2f bG = WG[kb * DD + n];
        v2f bM = WM[kb * DD + n];
        v2f bN = WN[kb * DD + n];
        cl0 = __builtin_amdgcn_wmma_f32_16x16x4_f32(false, ax0, false, bL, (short)0, cl0, false, false);
        cl1 = __builtin_amdgcn_wmma_f32_16x16x4_f32(false, ax1, false, bL, (short)0, cl1, false, false);
        cg0 = __builtin_amdgcn_wmma_f32_16x16x4_f32(false, ah0, false, bG, (short)0, cg0, false, false);
        cg1 = __builtin_amdgcn_wmma_f32_16x16x4_f32(false, ah1, false, bG, (short)0, cg1, false, false);
        cm0 = __builtin_amdgcn_wmma_f32_16x16x4_f32(false, ax0, false, bM, (short)0, cm0, false, false);
        cm1 = __builtin_amdgcn_wmma_f32_16x16x4_f32(false, ax1, false, bM, (short)0, cm1, false, false);
        cn0 = __builtin_amdgcn_wmma_f32_16x16x4_f32(false, ax0, false, bN, (short)0, cn0, false, false);
        cn1 = __builtin_amdgcn_wmma_f32_16x16x4_f32(false, ax1, false, bN, (short)0, cn1, false, false);
    }

    const float Bl = bl[n], Bg = bg[n], Bm = bm[n], Bn = bn[n];

    // Tile 0 (always valid: row0+15 <= 49999+)
    float* o0 = xout + (size_t)(row0 + 8 * kh) * DD + n;
#pragma unroll
    for (int i = 0; i < 8; ++i) {
        float v = cl0[i] + Bl + cg0[i] + Bg + (cm0[i] + Bm) * (cn0[i] + Bn);
        o0[(size_t)i * DD] = v > 0.f ? v : 0.f;
    }
    // Tile 1 (wave-uniform guard; only the final partial block skips it)
    if (row0 + 16 < NN) {
        float* o1 = xout + (size_t)(row0 + 16 + 8 * kh) * DD + n;
#pragma unroll
        for (int i = 0; i < 8; ++i) {
            float v = cl1[i] + Bl + cg1[i] + Bg + (cm1[i] + Bm) * (cn1[i] + Bn);
            o1[(size_t)i * DD] = v > 0.f ? v : 0.f;
        }
    }
}

// ---------------------------------------------------------------------------
// Inputs (setup_inputs order):
//  0 x  1 edge_index  2 edge_attr
//  3 fc11_w 4 fc11_b 5 fc12_w 6 fc12_b 7 fc13_w 8 fc13_b 9 conv1_w 10 conv1_b
// 11 fc21_w 12 fc21_b 13 fc22_w 14 fc22_b 15 fc23_w 16 fc23_b 17 conv2_w 18 conv2_b
// ---------------------------------------------------------------------------
extern "C" void kernel_launch(void* const* d_in, const int* in_sizes, int n_in,
                              void* d_out, int out_size, void* d_ws, size_t ws_size,
                              hipStream_t stream) {
    const float* x  = (const float*)d_in[0];
    const int*   ei = (const int*)d_in[1];
    const float* ea = (const float*)d_in[2];
    const float* fc11_w = (const float*)d_in[3];
    const float* fc11_b = (const float*)d_in[4];
    const float* fc12_w = (const float*)d_in[5];
    const float* fc12_b = (const float*)d_in[6];
    const float* fc13_w = (const float*)d_in[7];
    const float* fc13_b = (const float*)d_in[8];
    const float* conv1_w = (const float*)d_in[9];
    const float* conv1_b = (const float*)d_in[10];
    const float* fc21_w = (const float*)d_in[11];
    const float* fc21_b = (const float*)d_in[12];
    const float* fc22_w = (const float*)d_in[13];
    const float* fc22_b = (const float*)d_in[14];
    const float* fc23_w = (const float*)d_in[15];
    const float* fc23_b = (const float*)d_in[16];
    const float* conv2_w = (const float*)d_in[17];
    const float* conv2_b = (const float*)d_in[18];

    float* h    = (float*)d_ws;                  // layer-1 output  [N,D]
    float* hagg = h + (size_t)NN * DD;           // aggregation buf [N,D]
    float* wp   = hagg + (size_t)NN * DD;        // 8 packed weights, 16384 f each
    float* out  = (float*)d_out;

    const float* wsrc[8] = { fc11_w, conv1_w, fc12_w, fc13_w,
                             fc21_w, conv2_w, fc22_w, fc23_w };
    for (int i = 0; i < 8; ++i)
        pack_weight<<<dim3((64 * DD + 255) / 256), dim3(256), 0, stream>>>(
            wsrc[i], wp + (size_t)i * 2 * 64 * DD);

    float* p11 = wp + 0 * 16384;  float* pc1 = wp + 1 * 16384;
    float* p12 = wp + 2 * 16384;  float* p13 = wp + 3 * 16384;
    float* p21 = wp + 4 * 16384;  float* pc2 = wp + 5 * 16384;
    float* p22 = wp + 6 * 16384;  float* p23 = wp + 7 * 16384;

    const int n4 = NN * DD / 4;
    dim3 zg((n4 + 255) / 256), zb(256);
    dim3 sg((EE * 32 + 255) / 256), sb(256);     // 1 wave per edge
    dim3 gg((NN + 31) / 32), gb(256);            // 32 rows per block

    // ---- Layer 1 ----
    zero_f4<<<zg, zb, 0, stream>>>((float4*)hagg, n4);
    edge_scatter<<<sg, sb, 0, stream>>>(x, ei, ea, hagg);
    fused_layer<<<gg, gb, 0, stream>>>(x, hagg,
        p11, fc11_b, pc1, conv1_b, p12, fc12_b, p13, fc13_b, h);

    // ---- Layer 2 ----
    zero_f4<<<zg, zb, 0, stream>>>((float4*)hagg, n4);
    edge_scatter<<<sg, sb, 0, stream>>>(h, ei, ea, hagg);
    fused_layer<<<gg, gb, 0, stream>>>(h, hagg,
        p21, fc21_b, pc2, conv2_b, p22, fc22_b, p23, fc23_b, out);
}